// MoELayer_55473797595677
// MI455X (gfx1250) — compile-verified
//
#include <hip/hip_runtime.h>
#include <stdint.h>

// ---------------- problem constants ----------------
#define T_TOK 2048      // B*S tokens
#define HDIM  2048
#define EXP   64
#define TOPK  8
#define FDIM  512
#define CAP   320       // per-expert capacity
#define CPAD  336       // capacity padded to multiple of 16
#define F2    1024      // 2*F for shared experts

typedef float v2f __attribute__((ext_vector_type(2)));
typedef float v8f __attribute__((ext_vector_type(8)));
typedef int   v4i_vs __attribute__((vector_size(16)));   // matches builtin param type

// ---------------- CDNA5 async global->LDS path (guarded) ----------------
#if __has_builtin(__builtin_amdgcn_global_load_async_to_lds_b128) && \
    __has_builtin(__builtin_amdgcn_s_wait_asynccnt)
#define HAVE_ASYNC_LDS 1
#else
#define HAVE_ASYNC_LDS 0
#endif

__device__ __forceinline__ void ld_b128_to_lds(const float* g, float* l)
{
#if HAVE_ASYNC_LDS
    __builtin_amdgcn_global_load_async_to_lds_b128(
        (__attribute__((address_space(1))) v4i_vs*)(g),
        (__attribute__((address_space(3))) v4i_vs*)(l), 0, 0);
#else
    *(float4*)l = *(const float4*)g;
#endif
}

template <int N>
__device__ __forceinline__ void wait_async()
{
#if HAVE_ASYNC_LDS
    __builtin_amdgcn_s_wait_asynccnt(N);
#endif
}

// =====================================================================
// 1) Router: logits = x @ router_w ; softmax ; top-8 ; store probs
// =====================================================================
__global__ __launch_bounds__(256) void router_kernel(
    const float* __restrict__ x, const float* __restrict__ rw,
    float* __restrict__ probs, float* __restrict__ topv, int* __restrict__ topi)
{
    const int t = blockIdx.x;
    __shared__ float part[256];
    __shared__ float lg[EXP];

    const int e = threadIdx.x & 63;
    const int p = threadIdx.x >> 6;            // 0..3, H split in 4 chunks of 512
    const float* xr = x + (size_t)t * HDIM;

    float s = 0.f;
    const int h0 = p * 512;
    for (int h = h0; h < h0 + 512; ++h)
        s += xr[h] * rw[(size_t)h * EXP + e];
    part[threadIdx.x] = s;
    __syncthreads();

    if (threadIdx.x < EXP)
        lg[threadIdx.x] = part[threadIdx.x] + part[threadIdx.x + 64] +
                          part[threadIdx.x + 128] + part[threadIdx.x + 192];
    __syncthreads();

    if (threadIdx.x == 0) {
        float mx = lg[0];
        for (int i = 1; i < EXP; ++i) mx = lg[i] > mx ? lg[i] : mx;
        float sum = 0.f;
        for (int i = 0; i < EXP; ++i) { float v = __expf(lg[i] - mx); lg[i] = v; sum += v; }
        float inv = 1.f / sum;
        for (int i = 0; i < EXP; ++i) { lg[i] *= inv; probs[(size_t)t * EXP + i] = lg[i]; }
        // serial top-8, strict '>' => lowest index wins ties (matches lax.top_k)
        for (int k = 0; k < TOPK; ++k) {
            float bv = -1.f; int bi = 0;
            for (int i = 0; i < EXP; ++i)
                if (lg[i] > bv) { bv = lg[i]; bi = i; }
            topv[t * TOPK + k] = bv;
            topi[t * TOPK + k] = bi;
            lg[bi] = -1.f;
        }
    }
}

// =====================================================================
// 2) Dispatch: exact k-major capacity assignment (1 wave32 per expert)
// =====================================================================
__global__ __launch_bounds__(32) void dispatch_kernel(
    const int* __restrict__ topi, const float* __restrict__ topv,
    int* __restrict__ slots, float* __restrict__ wflat, int* __restrict__ cnt)
{
    const int e = blockIdx.x;
    const int lane = threadIdx.x;
    int count = 0;
    const unsigned lanemask_lt = (lane == 0) ? 0u : ((1u << lane) - 1u);
    for (int base = 0; base < TOPK * T_TOK; base += 32) {
        const int a = base + lane;
        const int t = a & (T_TOK - 1);
        const int k = a >> 11;                 // T=2048 => a / T
        const bool match = (topi[t * TOPK + k] == e);
        const unsigned mask = (unsigned)__ballot(match);   // wave32: low 32 bits
        if (match) {
            const int pos = count + __popc(mask & lanemask_lt);
            const bool keep = pos < CAP;
            slots[a] = keep ? pos : CAP;
            wflat[a] = keep ? topv[t * TOPK + k] : 0.f;
        }
        count += __popc(mask);
    }
    if (lane == 0) cnt[e] = count;
}

// =====================================================================
// 3) zero + scatter tokens to xd[E][CPAD][H]
// =====================================================================
__global__ void zero_kernel(float* __restrict__ p, size_t n)
{
    size_t i = (size_t)blockIdx.x * blockDim.x + threadIdx.x;
    const size_t stride = (size_t)gridDim.x * blockDim.x;
    for (; i < n; i += stride) p[i] = 0.f;
}

__global__ __launch_bounds__(256) void scatter_kernel(
    const float* __restrict__ xf, const int* __restrict__ topi,
    const int* __restrict__ slots, float* __restrict__ xd)
{
    const int a = blockIdx.x;
    const int slot = slots[a];
    if (slot >= CAP) return;                   // dropped -> stays zero
    const int t = a & (T_TOK - 1);
    const int k = a >> 11;
    const int e = topi[t * TOPK + k];
    const float* src = xf + (size_t)t * HDIM;
    float* dst = xd + ((size_t)e * CPAD + slot) * HDIM;
    for (int h = threadIdx.x; h < HDIM; h += 256) dst[h] = src[h];
}

// =====================================================================
// WMMA GEMM core (fp32, V_WMMA_F32_16X16X4_F32), async double-buffered.
// Block = 128 threads (4 waves), macro tile 64(M) x 64(N), K chunked by 32.
// Fragment layouts per CDNA5 ISA 7.12.2:
//   A 16x4 : m=lane&15, kh=(lane>>4)*2 ; v = {A[m,k0],A[m,k0+1]}
//   B 4x16 : n=lane&15 ; v = {B[kh,n], B[kh+1,n]}
//   C/D    : vgpr r -> row r + (lane>>4)*8, col lane&15
// NOTE: out-of-range M rows are loaded unpredicated (keeps ASYNCcnt
// bookkeeping wave-uniform); their garbage only reaches store-guarded rows.
// =====================================================================
#define BKC  32
#define APAD 36   // 8B-aligned, conflict-free A-fragment reads
#define BPAD 72   // disjoint bank sets for the two K-half lane groups

__device__ __forceinline__ v8f wmma_f32(v2f a, v2f b, v8f c)
{
    return __builtin_amdgcn_wmma_f32_16x16x4_f32(false, a, false, b, (short)0, c,
                                                 false, false);
}

// ---- fused dual GEMM: Out = relu(A@Bg) * (A@Bu) ----
__global__ __launch_bounds__(128) void dual_gemm_wmma(
    const float* __restrict__ A, int64_t ldA, int64_t strideAe,
    const float* __restrict__ Bg, const float* __restrict__ Bu,
    int64_t ldB, int64_t strideBe,
    float* __restrict__ O, int64_t ldO, int64_t strideOe,
    int M, int Kd)
{
    __shared__ __align__(16) float As [2][64][APAD];
    __shared__ __align__(16) float Bgs[2][BKC][BPAD];
    __shared__ __align__(16) float Bus[2][BKC][BPAD];

    const int e    = blockIdx.z;
    const int mblk = blockIdx.y;
    const int nblk = blockIdx.x;
    const int tid  = threadIdx.x;
    const int wv   = tid >> 5;
    const int lane = tid & 31;

    const float* Ae  = A  + (size_t)e * strideAe;
    const float* Bge = Bg + (size_t)e * strideBe;
    const float* Bue = Bu + (size_t)e * strideBe;
    float*       Oe  = O  + (size_t)e * strideOe;

    // 12 async instructions per wave per chunk (4 A + 4 Bg + 4 Bu)
    auto issue = [&](int kb, int buf) {
        #pragma unroll
        for (int q = 0; q < 4; ++q) {
            const int idx = tid + q * 128;          // 512 b128 xfers: 64r x 8q
            const int r  = idx >> 3;
            const int c4 = (idx & 7) * 4;
            ld_b128_to_lds(Ae + (size_t)(mblk * 64 + r) * ldA + kb + c4,
                           &As[buf][r][c4]);
        }
        #pragma unroll
        for (int q = 0; q < 4; ++q) {
            const int idx = tid + q * 128;          // 512 b128 xfers: 32r x 16q
            const int r  = idx >> 4;
            const int c4 = (idx & 15) * 4;
            const size_t off = (size_t)(kb + r) * ldB + nblk * 64 + c4;
            ld_b128_to_lds(Bge + off, &Bgs[buf][r][c4]);
            ld_b128_to_lds(Bue + off, &Bus[buf][r][c4]);
        }
    };

    v8f accg[4], accu[4];
    #pragma unroll
    for (int j = 0; j < 4; ++j) { accg[j] = (v8f){0,0,0,0,0,0,0,0}; accu[j] = (v8f){0,0,0,0,0,0,0,0}; }

    const int nchunk = Kd / BKC;
    issue(0, 0);
    const int m  = lane & 15;
    const int kh = (lane >> 4) * 2;

    for (int i = 0; i < nchunk; ++i) {
        const int cur = i & 1;
        if (i + 1 < nchunk) { issue((i + 1) * BKC, cur ^ 1); wait_async<12>(); }
        else                { wait_async<0>(); }
        __syncthreads();

        for (int kk = 0; kk < BKC; kk += 4) {
            v2f a = *(const v2f*)&As[cur][wv * 16 + m][kk + kh];
            #pragma unroll
            for (int j = 0; j < 4; ++j) {
                const int n = j * 16 + m;
                v2f bg; bg.x = Bgs[cur][kk + kh][n]; bg.y = Bgs[cur][kk + kh + 1][n];
                accg[j] = wmma_f32(a, bg, accg[j]);
                v2f bu; bu.x = Bus[cur][kk + kh][n]; bu.y = Bus[cur][kk + kh + 1][n];
                accu[j] = wmma_f32(a, bu, accu[j]);
            }
        }
        __syncthreads();
    }

    // epilogue: relu(g) * u
    #pragma unroll
    for (int j = 0; j < 4; ++j) {
        #pragma unroll
        for (int vr = 0; vr < 8; ++vr) {
            const int row  = wv * 16 + vr + ((lane >> 4) << 3);
            const int grow = mblk * 64 + row;
            if (grow < M) {
                float g = accg[j][vr]; g = g > 0.f ? g : 0.f;
                Oe[(size_t)grow * ldO + nblk * 64 + j * 16 + (lane & 15)] = g * accu[j][vr];
            }
        }
    }
}

// ---- plain GEMM: Out = A@B ----
__global__ __launch_bounds__(128) void gemm_wmma(
    const float* __restrict__ A, int64_t ldA, int64_t strideAe,
    const float* __restrict__ B, int64_t ldB, int64_t strideBe,
    float* __restrict__ O, int64_t ldO, int64_t strideOe,
    int M, int Kd)
{
    __shared__ __align__(16) float As[2][64][APAD];
    __shared__ __align__(16) float Bs[2][BKC][BPAD];

    const int e    = blockIdx.z;
    const int mblk = blockIdx.y;
    const int nblk = blockIdx.x;
    const int tid  = threadIdx.x;
    const int wv   = tid >> 5;
    const int lane = tid & 31;

    const float* Ae = A + (size_t)e * strideAe;
    const float* Be = B + (size_t)e * strideBe;
    float*       Oe = O + (size_t)e * strideOe;

    // 8 async instructions per wave per chunk (4 A + 4 B)
    auto issue = [&](int kb, int buf) {
        #pragma unroll
        for (int q = 0; q < 4; ++q) {
            const int idx = tid + q * 128;
            const int r  = idx >> 3;
            const int c4 = (idx & 7) * 4;
            ld_b128_to_lds(Ae + (size_t)(mblk * 64 + r) * ldA + kb + c4,
                           &As[buf][r][c4]);
        }
        #pragma unroll
        for (int q = 0; q < 4; ++q) {
            const int idx = tid + q * 128;
            const int r  = idx >> 4;
            const int c4 = (idx & 15) * 4;
            ld_b128_to_lds(Be + (size_t)(kb + r) * ldB + nblk * 64 + c4,
                           &Bs[buf][r][c4]);
        }
    };

    v8f acc[4];
    #pragma unroll
    for (int j = 0; j < 4; ++j) acc[j] = (v8f){0,0,0,0,0,0,0,0};

    const int nchunk = Kd / BKC;
    issue(0, 0);
    const int m  = lane & 15;
    const int kh = (lane >> 4) * 2;

    for (int i = 0; i < nchunk; ++i) {
        const int cur = i & 1;
        if (i + 1 < nchunk) { issue((i + 1) * BKC, cur ^ 1); wait_async<8>(); }
        else                { wait_async<0>(); }
        __syncthreads();

        for (int kk = 0; kk < BKC; kk += 4) {
            v2f a = *(const v2f*)&As[cur][wv * 16 + m][kk + kh];
            #pragma unroll
            for (int j = 0; j < 4; ++j) {
                const int n = j * 16 + m;
                v2f b; b.x = Bs[cur][kk + kh][n]; b.y = Bs[cur][kk + kh + 1][n];
                acc[j] = wmma_f32(a, b, acc[j]);
            }
        }
        __syncthreads();
    }

    #pragma unroll
    for (int j = 0; j < 4; ++j) {
        #pragma unroll
        for (int vr = 0; vr < 8; ++vr) {
            const int row  = wv * 16 + vr + ((lane >> 4) << 3);
            const int grow = mblk * 64 + row;
            if (grow < M)
                Oe[(size_t)grow * ldO + nblk * 64 + j * 16 + (lane & 15)] = acc[j][vr];
        }
    }
}

// =====================================================================
// 5) Combine: out[t] = shared_y[t] + sum_k w * y[e_k, slot_k]  (gather)
// =====================================================================
__global__ __launch_bounds__(256) void combine_kernel(
    const float* __restrict__ shared_y, const float* __restrict__ y,
    const int* __restrict__ topi, const int* __restrict__ slots,
    const float* __restrict__ wflat, float* __restrict__ out)
{
    const int t = blockIdx.x;
    __shared__ int   se[TOPK];
    __shared__ int   ss[TOPK];
    __shared__ float sw[TOPK];
    if (threadIdx.x < TOPK) {
        const int k = threadIdx.x;
        const int a = k * T_TOK + t;
        se[k] = topi[t * TOPK + k];
        ss[k] = slots[a];
        sw[k] = wflat[a];
    }
    __syncthreads();
    for (int h = threadIdx.x; h < HDIM; h += 256) {
        float accv = shared_y[(size_t)t * HDIM + h];
        #pragma unroll
        for (int k = 0; k < TOPK; ++k) {
            const float w = sw[k];
            if (w != 0.f)
                accv += w * y[((size_t)se[k] * CPAD + ss[k]) * HDIM + h];
        }
        out[(size_t)t * HDIM + h] = accv;
    }
}

// =====================================================================
// 6) aux loss
// =====================================================================
__global__ __launch_bounds__(256) void aux1_kernel(
    const float* __restrict__ probs, const int* __restrict__ cnt, float* __restrict__ term)
{
    const int e = blockIdx.x;
    __shared__ float red[256];
    float s = 0.f;
    for (int t = threadIdx.x; t < T_TOK; t += 256) s += probs[(size_t)t * EXP + e];
    red[threadIdx.x] = s;
    __syncthreads();
    for (int st = 128; st > 0; st >>= 1) {
        if (threadIdx.x < st) red[threadIdx.x] += red[threadIdx.x + st];
        __syncthreads();
    }
    if (threadIdx.x == 0) {
        const float mean_prob = red[0] / (float)T_TOK;
        const float frac = (float)cnt[e] / (float)(T_TOK * TOPK);
        term[e] = frac * mean_prob;
    }
}

__global__ __launch_bounds__(64) void aux2_kernel(const float* __restrict__ term,
                                                  float* __restrict__ out_aux)
{
    __shared__ float red[64];
    red[threadIdx.x] = term[threadIdx.x];
    __syncthreads();
    for (int st = 32; st > 0; st >>= 1) {
        if (threadIdx.x < st) red[threadIdx.x] += red[threadIdx.x + st];
        __syncthreads();
    }
    if (threadIdx.x == 0) out_aux[0] = (float)EXP * red[0];
}

// =====================================================================
// launch
// =====================================================================
extern "C" void kernel_launch(void* const* d_in, const int* in_sizes, int n_in,
                              void* d_out, int out_size, void* d_ws, size_t ws_size,
                              hipStream_t stream)
{
    (void)in_sizes; (void)n_in; (void)out_size; (void)ws_size;
    const float* x   = (const float*)d_in[0];
    const float* rw  = (const float*)d_in[1];
    const float* wg  = (const float*)d_in[2];
    const float* wu  = (const float*)d_in[3];
    const float* wd  = (const float*)d_in[4];
    const float* swg = (const float*)d_in[5];
    const float* swu = (const float*)d_in[6];
    const float* swd = (const float*)d_in[7];
    float* out = (float*)d_out;

    char* ws = (char*)d_ws;
    float* probs = (float*)(ws + 0);                      // T*E
    float* topv  = (float*)(ws + 524288);                 // T*K
    int*   topi  = (int*)  (ws + 589824);                 // T*K
    int*   slots = (int*)  (ws + 655360);                 // K*T
    float* wflat = (float*)(ws + 720896);                 // K*T
    int*   cnt   = (int*)  (ws + 786432);                 // E
    float* term  = (float*)(ws + 786688);                 // E
    const size_t xd_off   = 1048576;                              // 1 MB
    const size_t xd_sz    = (size_t)EXP * CPAD * HDIM * 4;        // 176,160,768
    const size_t act_off  = xd_off + xd_sz;
    const size_t act_sz   = (size_t)EXP * CPAD * FDIM * 4;        // 44,040,192
    const size_t acts_off = act_off + act_sz;
    const size_t acts_sz  = (size_t)T_TOK * 2 * F2 * 4;           // 16,777,216
    const size_t shy_off  = acts_off + acts_sz;
    float* xd    = (float*)(ws + xd_off);
    float* act   = (float*)(ws + act_off);
    float* act_s = (float*)(ws + acts_off);   // [T, 2*F2] column-concatenated
    float* shy   = (float*)(ws + shy_off);
    float* y     = xd;                        // reuse: xd dead after dual GEMM

    // 1) router
    router_kernel<<<T_TOK, 256, 0, stream>>>(x, rw, probs, topv, topi);
    // 2) dispatch
    dispatch_kernel<<<EXP, 32, 0, stream>>>(topi, topv, slots, wflat, cnt);
    // 3) zero + scatter
    zero_kernel<<<4096, 256, 0, stream>>>(xd, (size_t)EXP * CPAD * HDIM);
    scatter_kernel<<<TOPK * T_TOK, 256, 0, stream>>>(x, topi, slots, xd);
    // 4a) expert up/gate: act = relu(xd@wg)*(xd@wu)   [E,336,512]
    {
        dim3 g(FDIM / 64, (CPAD + 63) / 64, EXP);
        dual_gemm_wmma<<<g, 128, 0, stream>>>(
            xd, HDIM, (int64_t)CPAD * HDIM,
            wg, wu, FDIM, (int64_t)HDIM * FDIM,
            act, FDIM, (int64_t)CPAD * FDIM, CPAD, HDIM);
    }
    // 4b) expert down: y = act@wd   [E,336,2048]  (writes over xd)
    {
        dim3 g(HDIM / 64, (CPAD + 63) / 64, EXP);
        gemm_wmma<<<g, 128, 0, stream>>>(
            act, FDIM, (int64_t)CPAD * FDIM,
            wd, HDIM, (int64_t)FDIM * HDIM,
            y, HDIM, (int64_t)CPAD * HDIM, CPAD, FDIM);
    }
    // 4c) shared up/gate, 2 experts written into concatenated columns of act_s
    {
        dim3 g(F2 / 64, T_TOK / 64, 2);
        dual_gemm_wmma<<<g, 128, 0, stream>>>(
            x, HDIM, 0,
            swg, swu, F2, (int64_t)HDIM * F2,
            act_s, 2 * F2, F2 /* col offset per expert */, T_TOK, HDIM);
    }
    // 4d) shared down: shy = act_s[T,2048] @ swd_flat[2048,2048]
    {
        dim3 g(HDIM / 64, T_TOK / 64, 1);
        gemm_wmma<<<g, 128, 0, stream>>>(
            act_s, 2 * F2, 0,
            swd, HDIM, 0,
            shy, HDIM, 0, T_TOK, 2 * F2);
    }
    // 5) combine
    combine_kernel<<<T_TOK, 256, 0, stream>>>(shy, y, topi, slots, wflat, out);
    // 6) aux loss -> out[T*H]
    aux1_kernel<<<EXP, 256, 0, stream>>>(probs, cnt, term);
    aux2_kernel<<<1, 64, 0, stream>>>(term, out + (size_t)T_TOK * HDIM);
}